// GCN_44220983280301
// MI455X (gfx1250) — compile-verified
//
#include <hip/hip_runtime.h>
#include <hip/hip_bf16.h>

typedef float v2f __attribute__((ext_vector_type(2)));
typedef float v8f __attribute__((ext_vector_type(8)));

#define FEAT 64  // HID_DIM == OUT_DIM == 64

// ---------------------------------------------------------------------------
// Small elementwise helpers
// ---------------------------------------------------------------------------
__global__ void k_fill(float* __restrict__ p, float v, int n) {
  int i = blockIdx.x * blockDim.x + threadIdx.x;
  if (i < n) p[i] = v;
}

__global__ void k_deg_edges(const int* __restrict__ dst, const float* __restrict__ ew,
                            float* __restrict__ deg, int E) {
  int e = blockIdx.x * blockDim.x + threadIdx.x;
  if (e < E) unsafeAtomicAdd(&deg[dst[e]], ew[e]);
}

__global__ void k_dinv(float* __restrict__ deg_dinv, int n) {
  int i = blockIdx.x * blockDim.x + threadIdx.x;
  if (i < n) {
    float d = deg_dinv[i];
    deg_dinv[i] = (d > 0.0f) ? __frsqrt_rn(d) : 0.0f;
  }
}

// out[n][f] = bias[f] for n in [0,N)
__global__ void k_bias_init(float* __restrict__ out, const float* __restrict__ b, int total) {
  int i = blockIdx.x * blockDim.x + threadIdx.x;
  if (i < total) out[i] = b[i & (FEAT - 1)];
}

// ---------------------------------------------------------------------------
// GEMM via WMMA fp32: out[n][o] = sum_k act(in[n][k]) * W[o][k]
//   in : [nrows][K] row-major, W : [OUT][K] row-major, out : [nrows][OUT]
// One wave computes a 16-row x OUT tile.  Block = 256 threads = 8 waves.
// A fragment (16x4 f32): lane l (m = l&15, h = l>>4) holds A[m][kk+2h+v], v=0,1
// B fragment (4x16 f32): lane l (n = l&15, h = l>>4) holds B[kk+2h+v][n] = W[n][kk+2h+v]
// C/D (16x16 f32): vgpr r, lane l -> D[r + 8*h][l&15]
// ---------------------------------------------------------------------------
template <int K, bool RELU>
__global__ __launch_bounds__(256) void gemm_wmma(const float* __restrict__ in,
                                                 const float* __restrict__ W,
                                                 float* __restrict__ out, int nrows) {
  const int lane = threadIdx.x & 31;
  const int wave = threadIdx.x >> 5;
  const int row0 = (blockIdx.x * 8 + wave) * 16;
  if (row0 >= nrows) return;

  const int m = lane & 15;   // A row within tile / B column (output index)
  const int h = lane >> 4;   // lane half selects K sub-pair

  int r = row0 + m;
  if (r >= nrows) r = nrows - 1;  // keep EXEC all-1 for WMMA; stores are guarded
  const float* __restrict__ arow = in + (size_t)r * K;

  v8f acc0 = {}, acc1 = {}, acc2 = {}, acc3 = {};

  for (int kk = 0; kk < K; kk += 4) {
    v2f a = *(const v2f*)(arow + kk + 2 * h);
    if (RELU) {
      a.x = fmaxf(a.x, 0.0f);
      a.y = fmaxf(a.y, 0.0f);
    }
    const float* __restrict__ wk = W + kk + 2 * h;
    v2f b0 = *(const v2f*)(wk + (size_t)(0 * 16 + m) * K);
    v2f b1 = *(const v2f*)(wk + (size_t)(1 * 16 + m) * K);
    v2f b2 = *(const v2f*)(wk + (size_t)(2 * 16 + m) * K);
    v2f b3 = *(const v2f*)(wk + (size_t)(3 * 16 + m) * K);
    acc0 = __builtin_amdgcn_wmma_f32_16x16x4_f32(false, a, false, b0, (short)0, acc0, false, false);
    acc1 = __builtin_amdgcn_wmma_f32_16x16x4_f32(false, a, false, b1, (short)0, acc1, false, false);
    acc2 = __builtin_amdgcn_wmma_f32_16x16x4_f32(false, a, false, b2, (short)0, acc2, false, false);
    acc3 = __builtin_amdgcn_wmma_f32_16x16x4_f32(false, a, false, b3, (short)0, acc3, false, false);
  }

  // store D tiles: row = row0 + rr + 8*h, col = t*16 + m
  for (int rr = 0; rr < 8; ++rr) {
    int rowi = row0 + rr + 8 * h;
    if (rowi < nrows) {
      float* __restrict__ orow = out + (size_t)rowi * FEAT;
      orow[0 * 16 + m] = acc0[rr];
      orow[1 * 16 + m] = acc1[rr];
      orow[2 * 16 + m] = acc2[rr];
      orow[3 * 16 + m] = acc3[rr];
    }
  }
}

// ---------------------------------------------------------------------------
// Edge aggregation: out[dst] += h[src] * (dinv[src]*ew*dinv[dst])
// 64 threads per edge (feature-parallel): 256B coalesced gather + f32 atomics
// ---------------------------------------------------------------------------
__global__ void k_agg_edges(const int* __restrict__ src, const int* __restrict__ dst,
                            const float* __restrict__ ew, const float* __restrict__ dinv,
                            const float* __restrict__ h, float* __restrict__ out, int E) {
  int idx = blockIdx.x * blockDim.x + threadIdx.x;
  int e = idx >> 6;
  int f = idx & (FEAT - 1);
  if (e >= E) return;
  int s = src[e];
  int d = dst[e];
  const float* __restrict__ hrow = h + (size_t)s * FEAT;
  __builtin_prefetch(hrow, 0, 1);
  float norm = dinv[s] * ew[e] * dinv[d];
  unsafeAtomicAdd(&out[(size_t)d * FEAT + f], hrow[f] * norm);
}

// Self-loop contribution: out[i] += h[i] * dinv[i]^2  (runs after edge kernel,
// stream-serialized -> plain read-modify-write, no atomics needed)
__global__ void k_agg_self(const float* __restrict__ dinv, const float* __restrict__ h,
                           float* __restrict__ out, int total) {
  int idx = blockIdx.x * blockDim.x + threadIdx.x;
  if (idx >= total) return;
  int i = idx >> 6;
  float w = dinv[i] * dinv[i];
  out[idx] += h[idx] * w;
}

// ---------------------------------------------------------------------------
// Host-side orchestration
// ---------------------------------------------------------------------------
extern "C" void kernel_launch(void* const* d_in, const int* in_sizes, int n_in,
                              void* d_out, int out_size, void* d_ws, size_t ws_size,
                              hipStream_t stream) {
  const float* x  = (const float*)d_in[0];        // [N,128]
  const int*   ei = (const int*)d_in[1];          // [2,E]
  const float* ew = (const float*)d_in[2];        // [E]
  const float* W1 = (const float*)d_in[3];        // [64,128]
  const float* b1 = (const float*)d_in[4];        // [64]
  const float* W2 = (const float*)d_in[5];        // [64,64]
  const float* b2 = (const float*)d_in[6];        // [64]

  const int N = in_sizes[0] / 128;
  const int E = in_sizes[2];
  const int* src = ei;
  const int* dst = ei + E;

  // workspace layout: dinv[N] | bufA[64N] | bufB[64N]   (~51.6 MB for N=100k)
  float* dinv = (float*)d_ws;
  float* bufA = dinv + N;            // h1, later h2
  float* bufB = bufA + (size_t)N * FEAT;  // agg1

  const int T = 256;
  dim3 blk(T);
  dim3 gN((N + T - 1) / T);
  dim3 gE((E + T - 1) / T);
  dim3 gNF(((size_t)N * FEAT + T - 1) / T);
  dim3 gEF(((size_t)E * FEAT + T - 1) / T);
  dim3 gGemm((N + 127) / 128);  // 8 waves/block, 16 rows/wave

  // ---- normalization: deg = 1 (self loop) + sum_in(ew); dinv = rsqrt(deg)
  k_fill<<<gN, blk, 0, stream>>>(dinv, 1.0f, N);
  k_deg_edges<<<gE, blk, 0, stream>>>(dst, ew, dinv, E);
  k_dinv<<<gN, blk, 0, stream>>>(dinv, N);

  // ---- layer 1: h1 = x @ W1^T
  gemm_wmma<128, false><<<gGemm, blk, 0, stream>>>(x, W1, bufA, N);
  // agg1 = scatter(h1*norm) + b1
  k_bias_init<<<gNF, blk, 0, stream>>>(bufB, b1, N * FEAT);
  k_agg_edges<<<gEF, blk, 0, stream>>>(src, dst, ew, dinv, bufA, bufB, E);
  k_agg_self<<<gNF, blk, 0, stream>>>(dinv, bufA, bufB, N * FEAT);

  // ---- layer 2: h2 = relu(agg1) @ W2^T   (ReLU fused into A-fragment load)
  gemm_wmma<64, true><<<gGemm, blk, 0, stream>>>(bufB, W2, bufA, N);
  // out = scatter(h2*norm) + b2
  float* out = (float*)d_out;
  k_bias_init<<<gNF, blk, 0, stream>>>(out, b2, N * FEAT);
  k_agg_edges<<<gEF, blk, 0, stream>>>(src, dst, ew, dinv, bufA, out, E);
  k_agg_self<<<gNF, blk, 0, stream>>>(dinv, bufA, out, N * FEAT);
}